// GINELayer_20005957664841
// MI455X (gfx1250) — compile-verified
//
#include <hip/hip_runtime.h>

typedef __attribute__((ext_vector_type(2))) float v2f;
typedef __attribute__((ext_vector_type(8))) float v8f;

#define D 128
#define RT 4                 // row tiles per block
#define ROWS (RT * 16)       // 64 rows per block

// ---------------------------------------------------------------------------
// Kernel 1: h_pre = 2 * node_feat   (also initializes poisoned scratch)
// ---------------------------------------------------------------------------
__global__ void gine_init(const float* __restrict__ node,
                          float* __restrict__ hpre, int n4) {
  int i = blockIdx.x * blockDim.x + threadIdx.x;
  if (i < n4) {
    float4 v = ((const float4*)node)[i];
    ((float4*)hpre)[i] = make_float4(2.f * v.x, 2.f * v.y, 2.f * v.z, 2.f * v.w);
  }
}

// ---------------------------------------------------------------------------
// Kernel 2: for each edge e: h_pre[dst[e]] += node[src[e]] + edge[e]
// One wave (32 lanes) per edge, 4 features per lane (float4 loads),
// native f32 global atomics (no CAS loop).
// ---------------------------------------------------------------------------
__global__ __launch_bounds__(256) void gine_scatter(
    const float* __restrict__ node, const float* __restrict__ edge,
    const int* __restrict__ src, const int* __restrict__ dst,
    float* __restrict__ hpre, int nEdges) {
  int e = blockIdx.x * 8 + (threadIdx.x >> 5);
  if (e >= nEdges) return;
  int lane = threadIdx.x & 31;
  int f = lane * 4;
  int s = src[e];
  int d = dst[e];
  float4 a = *(const float4*)(node + (size_t)s * D + f);
  float4 b = *(const float4*)(edge + (size_t)e * D + f);
  float* p = hpre + (size_t)d * D + f;
  unsafeAtomicAdd(p + 0, a.x + b.x);
  unsafeAtomicAdd(p + 1, a.y + b.y);
  unsafeAtomicAdd(p + 2, a.z + b.z);
  unsafeAtomicAdd(p + 3, a.w + b.w);
}

// ---------------------------------------------------------------------------
// Kernel 3: fused MLP:
//   h1  = LeakyReLU(h_pre @ W1 + b1)
//   out = h1 @ W2 + b2
// One block per 64 rows (4 row-tiles); 8 waves, each wave owns a 16-col tile
// and 4 row-tiles -> each B (weight) fragment is reused 4x in registers.
// A tile staged memory->LDS via CDNA5 GLOBAL_LOAD_ASYNC_TO_LDS_B128
// (ASYNCcnt path), single 32KB LDS buffer reused for the activated h1 tile.
// W1/W2 (64 KB each) stay hot in L2 (192 MB) across all blocks.
// ---------------------------------------------------------------------------
__global__ __launch_bounds__(256) void gine_mlp(
    const float* __restrict__ hpre,
    const float* __restrict__ W1, const float* __restrict__ b1,
    const float* __restrict__ W2, const float* __restrict__ b2,
    float* __restrict__ out, int nNodes) {
  __shared__ float lds[ROWS * D];  // 64x128 f32 = 32 KB

  const int row0  = blockIdx.x * ROWS;
  const int wave  = threadIdx.x >> 5;   // 0..7 -> 16-col tile
  const int lane  = threadIdx.x & 31;
  const int m     = lane & 15;          // row within tile (A), col within tile (B/C)
  const int khalf = lane >> 4;          // 0: K,K+1 ; 1: K+2,K+3  (A/B frag halves)
  const int col   = wave * 16 + m;      // output column 0..127

  // ---- stage 64x128 A tile: global -> LDS directly (async, ASYNCcnt) ----
  // 8 passes x 256 lanes x 16B = 32 KB. OOB rows clamped (outputs guarded).
  {
    const int nv4 = ROWS * D / 4;  // 2048 float4
#pragma unroll
    for (int p = 0; p < nv4 / 256; ++p) {
      int idx4 = p * 256 + threadIdx.x;      // float4 index in tile
      int r    = idx4 / (D / 4);             // tile row
      int c4   = idx4 % (D / 4);             // float4 col
      int gr   = row0 + r;
      if (gr > nNodes - 1) gr = nNodes - 1;  // clamp (rows never mix in GEMM)
      unsigned ldsoff = (unsigned)(uintptr_t)&lds[idx4 * 4];
      unsigned goff   = (unsigned)(((unsigned)gr * D + c4 * 4) * 4u); // bytes
      // GVS mode: mem_addr = SGPR base + VGPR i32 offset
      asm volatile("global_load_async_to_lds_b128 %0, %1, %2"
                   :: "v"(ldsoff), "v"(goff), "s"(hpre)
                   : "memory");
    }
    asm volatile("s_wait_asynccnt 0" ::: "memory");
  }
  __syncthreads();

  // ---- GEMM1: acc[t](16x16) = A_t(16x128) @ W1[:, col tile] ----
  v8f acc[RT];
#pragma unroll
  for (int t = 0; t < RT; ++t) acc[t] = (v8f){};

  for (int k = 0; k < D; k += 4) {
    int ka = k + 2 * khalf;
    v2f b;
    b[0] = W1[(size_t)ka * D + col];
    b[1] = W1[(size_t)(ka + 1) * D + col];
#pragma unroll
    for (int t = 0; t < RT; ++t) {
      v2f a;
      a[0] = lds[(t * 16 + m) * D + ka];
      a[1] = lds[(t * 16 + m) * D + ka + 1];
      acc[t] = __builtin_amdgcn_wmma_f32_16x16x4_f32(false, a, false, b,
                                                     (short)0, acc[t], false, false);
    }
  }
  __syncthreads();  // everyone done reading the A tile

  // ---- bias + LeakyReLU(0.01); scatter C-layout into LDS row-major ----
  float bias1 = b1[col];
#pragma unroll
  for (int t = 0; t < RT; ++t) {
#pragma unroll
    for (int v = 0; v < 8; ++v) {
      float x = acc[t][v] + bias1;
      x = x > 0.f ? x : 0.01f * x;
      int mm = t * 16 + v + 8 * khalf;   // C layout: M = v + 8*(lane>=16)
      lds[mm * D + col] = x;
    }
  }
  __syncthreads();

  // ---- GEMM2: acc2[t] = h1_t(16x128) @ W2[:, col tile] ----
  v8f acc2[RT];
#pragma unroll
  for (int t = 0; t < RT; ++t) acc2[t] = (v8f){};

  for (int k = 0; k < D; k += 4) {
    int ka = k + 2 * khalf;
    v2f b;
    b[0] = W2[(size_t)ka * D + col];
    b[1] = W2[(size_t)(ka + 1) * D + col];
#pragma unroll
    for (int t = 0; t < RT; ++t) {
      v2f a;
      a[0] = lds[(t * 16 + m) * D + ka];
      a[1] = lds[(t * 16 + m) * D + ka + 1];
      acc2[t] = __builtin_amdgcn_wmma_f32_16x16x4_f32(false, a, false, b,
                                                      (short)0, acc2[t], false, false);
    }
  }

  // ---- bias + store ----
  float bias2 = b2[col];
#pragma unroll
  for (int t = 0; t < RT; ++t) {
#pragma unroll
    for (int v = 0; v < 8; ++v) {
      int mm = t * 16 + v + 8 * khalf;
      int gr = row0 + mm;
      if (gr < nNodes) out[(size_t)gr * D + col] = acc2[t][v] + bias2;
    }
  }
}

// ---------------------------------------------------------------------------
extern "C" void kernel_launch(void* const* d_in, const int* in_sizes, int n_in,
                              void* d_out, int out_size, void* d_ws, size_t ws_size,
                              hipStream_t stream) {
  const float* node = (const float*)d_in[0];
  const float* edge = (const float*)d_in[1];
  const int*   src  = (const int*)d_in[2];
  const int*   dst  = (const int*)d_in[3];
  const float* W1   = (const float*)d_in[4];
  const float* b1   = (const float*)d_in[5];
  const float* W2   = (const float*)d_in[6];
  const float* b2   = (const float*)d_in[7];
  float* out = (float*)d_out;

  const int nNodes = in_sizes[0] / D;   // 100000
  const int nEdges = in_sizes[2];       // 600000
  float* hpre = (float*)d_ws;           // N*D floats scratch

  // h_pre = 2 * node_feat
  int n4 = nNodes * D / 4;
  gine_init<<<(n4 + 255) / 256, 256, 0, stream>>>(node, hpre, n4);

  // scatter edges (8 edges per 256-thread block)
  gine_scatter<<<(nEdges + 7) / 8, 256, 0, stream>>>(node, edge, src, dst,
                                                     hpre, nEdges);

  // fused 2-layer MLP with WMMA (64 rows per block)
  int nblk = (nNodes + ROWS - 1) / ROWS;
  gine_mlp<<<nblk, 256, 0, stream>>>(hpre, W1, b1, W2, b2, out, nNodes);
}